// net_28613072126490
// MI455X (gfx1250) — compile-verified
//
#include <hip/hip_runtime.h>
#include <hip/hip_bf16.h>

// ---------------------------------------------------------------------------
// GAT-style edge attention on MI455X (gfx1250):
//   qk = x @ W^T + b        WMMA f32 16x16x4 GEMM, 64x64 macro-tile per wave
//                           (16 v_wmma per 8 global_load_b64 in the K-loop;
//                            ~0.9GB L2 traffic; 104MB qk table is L2-resident)
//   scores[e,h] = <q[src,h,:], k[dst,h,:]>   wave-per-edge L2 gather
//   segment softmax over src  (encoded atomic u32 max, f32 atomic add)
//   out[e] = mean_h attn[e,h]
// ---------------------------------------------------------------------------

typedef __attribute__((ext_vector_type(2))) float v2f;
typedef __attribute__((ext_vector_type(8))) float v8f;

// Order-preserving float<->uint encoding so segment-max can use native
// global_atomic_max_u32 (handles negative scores; identity element is 0).
__device__ __forceinline__ unsigned fenc(float f) {
    unsigned u = __float_as_uint(f);
    return (u & 0x80000000u) ? ~u : (u | 0x80000000u);
}
__device__ __forceinline__ float fdec(unsigned e) {
    return (e & 0x80000000u) ? __uint_as_float(e & 0x7FFFFFFFu)
                             : __uint_as_float(~e);
}

// ---------------------------------------------------------------------------
// K0: zero init for smax (encoded) + denom (contiguous region)
// ---------------------------------------------------------------------------
__global__ void k0_zero(unsigned* __restrict__ p, size_t n) {
    size_t i = (size_t)blockIdx.x * blockDim.x + threadIdx.x;
    if (i < n) p[i] = 0u;
}

// ---------------------------------------------------------------------------
// K1: qk[N x C] = x[N x F] @ W[C x F]^T + b[C]
// One wave32 computes a 64x64 output macro-tile = 4x4 WMMA 16x16 sub-tiles.
// Per K-step (K+=4): 4 A loads + 4 B loads (b64) feed 16 v_wmma_f32_16x16x4.
// A 16x4 layout: lane(M=lane&15) holds K = (lane>>4)*2 + {0,1} in {v0,v1}.
// B 4x16 layout: lane(N=lane&15) holds K = (lane>>4)*2 + {0,1} in {v0,v1},
//                B[k][n] = W^T[k][n] = W[n][k].
// C/D layout:   VGPR r = D[M = r + (lane>=16 ? 8 : 0)][N = lane&15].
// Out-of-range rows/cols: addresses are CLAMPED (no fault) — garbage values
// only pollute rows/cols of D that are never stored, so no zeroing needed.
// EXEC stays all-ones: blockDim=32, no divergent control flow before WMMA.
// ---------------------------------------------------------------------------
__global__ __launch_bounds__(32)
void k1_gemm_wmma(const float* __restrict__ x, const float* __restrict__ W,
                  const float* __restrict__ bias, float* __restrict__ qk,
                  int N, int F, int C) {
    const int lane = threadIdx.x;             // 0..31 (wave32)
    const int M0   = blockIdx.x * 64;         // row macro-tile base
    const int N0   = blockIdx.y * 64;         // col macro-tile base
    const int lm   = lane & 15;
    const int kh   = (lane >> 4) * 2;         // 0 or 2

    // Clamped per-sub-tile row pointers (A from x, B from W)
    const float* __restrict__ xrow[4];
    const float* __restrict__ wrow[4];
    #pragma unroll
    for (int mi = 0; mi < 4; ++mi) {
        int r = M0 + mi * 16 + lm;
        xrow[mi] = x + (size_t)(r < N ? r : N - 1) * F;
    }
    #pragma unroll
    for (int ni = 0; ni < 4; ++ni) {
        int c = N0 + ni * 16 + lm;
        wrow[ni] = W + (size_t)(c < C ? c : C - 1) * F;
    }

    v8f acc[4][4];
    #pragma unroll
    for (int mi = 0; mi < 4; ++mi)
        #pragma unroll
        for (int ni = 0; ni < 4; ++ni)
            acc[mi][ni] = (v8f){};

    for (int k = 0; k < F; k += 4) {
        v2f a[4], b[4];
        #pragma unroll
        for (int mi = 0; mi < 4; ++mi) {
            a[mi].x = xrow[mi][k + kh];
            a[mi].y = xrow[mi][k + kh + 1];
        }
        #pragma unroll
        for (int ni = 0; ni < 4; ++ni) {
            b[ni].x = wrow[ni][k + kh];
            b[ni].y = wrow[ni][k + kh + 1];
        }
        #pragma unroll
        for (int mi = 0; mi < 4; ++mi)
            #pragma unroll
            for (int ni = 0; ni < 4; ++ni)
                acc[mi][ni] = __builtin_amdgcn_wmma_f32_16x16x4_f32(
                    /*neg_a=*/false, a[mi], /*neg_b=*/false, b[ni],
                    /*c_mod=*/(short)0, acc[mi][ni],
                    /*reuse_a=*/false, /*reuse_b=*/false);
    }

    // Store with bounds guards; bias fused.
    const int rsub = (lane >> 4) << 3;        // 0 or 8
    #pragma unroll
    for (int ni = 0; ni < 4; ++ni) {
        const int col = N0 + ni * 16 + lm;
        if (col >= C) continue;
        const float bv = bias[col];
        #pragma unroll
        for (int mi = 0; mi < 4; ++mi) {
            const int rbase = M0 + mi * 16 + rsub;
            #pragma unroll
            for (int r = 0; r < 8; ++r) {
                const int row = rbase + r;
                if (row < N)
                    qk[(size_t)row * C + col] = acc[mi][ni][r] + bv;
            }
        }
    }
}

// ---------------------------------------------------------------------------
// K2: one wave32 per edge. Coalesced gather of the src q-row / dst k-row
// (L2-resident), per-head dot with butterfly reduce, then score store +
// encoded atomic segment-max.
// ---------------------------------------------------------------------------
__global__ __launch_bounds__(256)
void k2_scores(const float* __restrict__ qk, const int* __restrict__ ei,
               float* __restrict__ scores, unsigned* __restrict__ smax,
               int E, int C, int H, int P) {
    const int lane = threadIdx.x & 31;
    const int e = blockIdx.x * (blockDim.x >> 5) + (threadIdx.x >> 5);
    if (e >= E) return;

    const int s = ei[e];
    const int d = ei[E + e];
    const float* __restrict__ srow = qk + (size_t)s * C;  // q half per head
    const float* __restrict__ drow = qk + (size_t)d * C;  // k half per head

    for (int h = 0; h < H; ++h) {
        const float* __restrict__ qh = srow + h * 2 * P;       // q[s,h,0..P)
        const float* __restrict__ kh = drow + h * 2 * P + P;   // k[d,h,0..P)
        float a = 0.0f;
        for (int p = lane; p < P; p += 32)
            a += qh[p] * kh[p];
        #pragma unroll
        for (int off = 16; off > 0; off >>= 1)
            a += __shfl_xor(a, off, 32);
        if (lane == 0) {
            scores[(size_t)e * H + h] = a;
            atomicMax(&smax[(size_t)s * H + h], fenc(a));
        }
    }
}

// ---------------------------------------------------------------------------
// K3: ex = exp(score - smax[src]); in-place overwrite; segment-sum via
// global_atomic_add_f32 into denom.
// ---------------------------------------------------------------------------
__global__ __launch_bounds__(256)
void k3_exp_denom(float* __restrict__ scores, const unsigned* __restrict__ smax,
                  const int* __restrict__ ei, float* __restrict__ denom,
                  int E, int H) {
    size_t i = (size_t)blockIdx.x * blockDim.x + threadIdx.x;
    if (i >= (size_t)E * H) return;
    const int e = (int)(i / H);
    const int h = (int)(i % H);
    const int s = ei[e];
    const float m  = fdec(smax[(size_t)s * H + h]);
    const float ex = __expf(scores[i] - m);
    scores[i] = ex;
    atomicAdd(&denom[(size_t)s * H + h], ex);
}

// ---------------------------------------------------------------------------
// K4: out[e] = mean_h( ex[e,h] / denom[src,h] )
// ---------------------------------------------------------------------------
__global__ __launch_bounds__(256)
void k4_out(const float* __restrict__ ex, const float* __restrict__ denom,
            const int* __restrict__ ei, float* __restrict__ out, int E, int H) {
    const int e = blockIdx.x * blockDim.x + threadIdx.x;
    if (e >= E) return;
    const int s = ei[e];
    float sum = 0.0f;
    for (int h = 0; h < H; ++h)
        sum += ex[(size_t)e * H + h] / denom[(size_t)s * H + h];
    out[e] = sum / (float)H;
}

// ---------------------------------------------------------------------------
extern "C" void kernel_launch(void* const* d_in, const int* in_sizes, int n_in,
                              void* d_out, int out_size, void* d_ws, size_t ws_size,
                              hipStream_t stream) {
    const float* x    = (const float*)d_in[0];
    const int*   ei   = (const int*)d_in[1];
    const float* W    = (const float*)d_in[2];
    const float* bias = (const float*)d_in[3];
    // d_in[4] = num_heads lives on device; derive all shapes from sizes:
    const int C = in_sizes[3];           // 2*dk (= 520)
    const int F = in_sizes[2] / C;       // 256
    const int N = in_sizes[0] / F;       // 50000
    const int E = in_sizes[1] / 2;       // 800000
    const int H = (C - 2 * F) / 2;       // 4   (2*(F/H+1)*H = 2F + 2H = C)
    const int P = F / H + 1;             // 65

    // workspace layout
    float*    qk     = (float*)d_ws;                               // N*C
    float*    scores = qk + (size_t)N * C;                         // E*H
    unsigned* smax   = (unsigned*)(scores + (size_t)E * H);        // N*H
    float*    denom  = (float*)(smax + (size_t)N * H);             // N*H

    // K0: zero smax + denom (contiguous 2*N*H words)
    {
        size_t n = (size_t)2 * N * H;
        k0_zero<<<(unsigned)((n + 255) / 256), 256, 0, stream>>>(smax, n);
    }

    // K1: WMMA GEMM, 64x64 macro-tile per wave  (782 x 9 tiles)
    {
        dim3 grid((N + 63) / 64, (C + 63) / 64);
        k1_gemm_wmma<<<grid, 32, 0, stream>>>(x, W, bias, qk, N, F, C);
    }

    // K2: per-edge scores + segment max (8 edges per 256-thread block)
    {
        const int epb = 256 / 32;
        k2_scores<<<(E + epb - 1) / epb, 256, 0, stream>>>(qk, ei, scores, smax,
                                                           E, C, H, P);
    }

    // K3: exp + segment sum
    {
        size_t n = (size_t)E * H;
        k3_exp_denom<<<(unsigned)((n + 255) / 256), 256, 0, stream>>>(scores, smax,
                                                                      ei, denom, E, H);
    }

    // K4: normalize + head mean
    k4_out<<<(E + 255) / 256, 256, 0, stream>>>(scores, denom, ei,
                                                (float*)d_out, E, H);
}